// FFD_4638564680323
// MI455X (gfx1250) — compile-verified
//
#include <hip/hip_runtime.h>

// FFD cubic B-spline evaluation (disp + Jacobian) for MI455X / gfx1250.
// Whole 19x19x19x3 control grid (82,308 B) is DMA'd into LDS with a single
// Tensor Data Mover descriptor (tensor_load_to_lds + s_wait_tensorcnt), then
// each thread evaluates one point via separable contractions gathering from
// LDS. Input domain guarantees base indices in [1,16], so the 4x4x4 stencil
// never clamps; z-neighbors are 12 contiguous LDS floats -> merged ds_load2.

#define GD      19
#define GSIZE   (GD * GD * GD * 3)      // 20577 floats = 82308 bytes
#define ORG     (-144.0f)               // ORIGIN per axis
#define INV_SP  (0.0625f)               // 1/16 spacing

typedef unsigned int        u32;
typedef unsigned long long  u64;
typedef u32 __attribute__((ext_vector_type(4))) v4u;
typedef int __attribute__((ext_vector_type(4))) v4i;
typedef int __attribute__((ext_vector_type(8))) v8i;

__device__ __forceinline__ void bsw(float t, float B[4], float dB[4]) {
    float t2  = t * t;
    float t3  = t2 * t;
    float omt = 1.0f - t;
    B[0]  = omt * omt * omt * (1.0f / 6.0f);
    B[1]  = (3.0f * t3 - 6.0f * t2 + 4.0f) * (1.0f / 6.0f);
    B[2]  = (-3.0f * t3 + 3.0f * t2 + 3.0f * t + 1.0f) * (1.0f / 6.0f);
    B[3]  = t3 * (1.0f / 6.0f);
    // derivative, pre-scaled by 1/spacing (Jacobian divides by spacing)
    dB[0] = -0.5f * omt * omt * INV_SP;
    dB[1] = (3.0f * t2 - 4.0f * t) * 0.5f * INV_SP;
    dB[2] = (-3.0f * t2 + 2.0f * t + 1.0f) * 0.5f * INV_SP;
    dB[3] = 0.5f * t2 * INV_SP;
}

__global__ __launch_bounds__(256) void ffd_eval_kernel(
        const float* __restrict__ points,
        const float* __restrict__ grid,
        float* __restrict__ out,
        int n)
{
    __shared__ float sgrid[GSIZE];

#if __has_builtin(__builtin_amdgcn_tensor_load_to_lds)
    // One TDM descriptor moves the entire control grid into LDS (TENSORcnt).
    if (threadIdx.x < 32) {   // wave 0 only: one DMA issue per workgroup
        u64 ga      = (u64)(const void*)grid;
        u32 lds_off = (u32)(u64)(void*)&sgrid[0];   // flat addr low 32b == LDS byte offset

        v4u g0;
        g0.x = 1u;                                        // count=1 (valid), user mode
        g0.y = lds_off;                                   // lds_addr (bytes)
        g0.z = (u32)(ga & 0xFFFFFFFFu);                   // global_addr[31:0]
        g0.w = (u32)((ga >> 32) & 0x01FFFFFFu)            // global_addr[56:32]
             | (2u << 30);                                // type = 2 ("image")

        v8i g1;
        g1[0] = (int)(2u << 16);                          // data_size=4B; wg_mask=0; no pad/iter
        g1[1] = (int)((u32)(GSIZE & 0xFFFFu) << 16);      // tensor_dim0[15:0]
        g1[2] = (int)(((u32)GSIZE >> 16) | (1u << 16));   // tensor_dim0[31:16]; tensor_dim1=1
        g1[3] = (int)((u32)(GSIZE & 0xFFFFu) << 16);      // tile_dim0 = 20577 (one full line)
        g1[4] = 1;                                        // tile_dim1=1, tile_dim2 unused
        g1[5] = (int)GSIZE;                               // tensor_dim0_stride[31:0]
        g1[6] = 0;
        g1[7] = 0;
        v4i gz4 = {0, 0, 0, 0};                           // groups 2/3: unused dims
        v8i gz8 = {0, 0, 0, 0, 0, 0, 0, 0};

        __builtin_amdgcn_tensor_load_to_lds(g0, g1, gz4, gz4, gz8, 0);
        __builtin_amdgcn_s_wait_tensorcnt(0);
    }
    __syncthreads();
#else
    for (int i = threadIdx.x; i < GSIZE; i += blockDim.x) sgrid[i] = grid[i];
    __syncthreads();
#endif

    const int stride = gridDim.x * blockDim.x;
    for (int p = blockIdx.x * blockDim.x + threadIdx.x; p < n; p += stride) {
        float px = points[p * 3 + 0];
        float py = points[p * 3 + 1];
        float pz = points[p * 3 + 2];

        float ux = (px - ORG) * INV_SP;
        float uy = (py - ORG) * INV_SP;
        float uz = (pz - ORG) * INV_SP;
        float fx = floorf(ux), fy = floorf(uy), fz = floorf(uz);
        float tx = ux - fx,    ty = uy - fy,    tz = uz - fz;
        // Stencil base (points span [-120,120) -> base in [1,16], never clamps;
        // LDS OOB loads return 0 by hardware rule, so this is also fault-safe.
        int bx = (int)fx - 1;
        int by = (int)fy - 1;
        int bz = (int)fz - 1;

        float Bx[4], dBx[4], By[4], dBy[4], Bz[4], dBz[4];
        bsw(tx, Bx, dBx);
        bsw(ty, By, dBy);
        bsw(tz, Bz, dBz);

        float D[3]  = {0.f, 0.f, 0.f};   // displacement   (Bx By Bz)
        float JX[3] = {0.f, 0.f, 0.f};   // d/dx           (dBx By Bz)
        float JY[3] = {0.f, 0.f, 0.f};   // d/dy           (Bx dBy Bz)
        float JZ[3] = {0.f, 0.f, 0.f};   // d/dz           (Bx By dBz)

#pragma unroll
        for (int i = 0; i < 4; ++i) {
            int rowx = (bx + i) * GD;
            float sBB[3] = {0.f, 0.f, 0.f};   // sum_j By  * (sum_k Bz  V)
            float sdB[3] = {0.f, 0.f, 0.f};   // sum_j dBy * (sum_k Bz  V)
            float sBd[3] = {0.f, 0.f, 0.f};   // sum_j By  * (sum_k dBz V)
#pragma unroll
            for (int j = 0; j < 4; ++j) {
                // 4 z-neighbors = 12 contiguous floats, single shared base
                const float* v = &sgrid[((rowx + by + j) * GD + bz) * 3];
                float t0[3] = {0.f, 0.f, 0.f};
                float t1[3] = {0.f, 0.f, 0.f};
#pragma unroll
                for (int k = 0; k < 4; ++k) {
                    float wz = Bz[k], dwz = dBz[k];
#pragma unroll
                    for (int c = 0; c < 3; ++c) {
                        float vv = v[k * 3 + c];
                        t0[c] = fmaf(wz,  vv, t0[c]);
                        t1[c] = fmaf(dwz, vv, t1[c]);
                    }
                }
                float wy = By[j], dwy = dBy[j];
#pragma unroll
                for (int c = 0; c < 3; ++c) {
                    sBB[c] = fmaf(wy,  t0[c], sBB[c]);
                    sdB[c] = fmaf(dwy, t0[c], sdB[c]);
                    sBd[c] = fmaf(wy,  t1[c], sBd[c]);
                }
            }
            float wx = Bx[i], dwx = dBx[i];
#pragma unroll
            for (int c = 0; c < 3; ++c) {
                D[c]  = fmaf(wx,  sBB[c], D[c]);
                JX[c] = fmaf(dwx, sBB[c], JX[c]);
                JY[c] = fmaf(wx,  sdB[c], JY[c]);
                JZ[c] = fmaf(wx,  sBd[c], JZ[c]);
            }
        }

        // out layout: disp [n,3] flat, then jac [n,3,3] flat (stack axis=-1 + eye)
        float* dout = out + (size_t)p * 3;
        dout[0] = D[0]; dout[1] = D[1]; dout[2] = D[2];

        float* jout = out + (size_t)n * 3 + (size_t)p * 9;
#pragma unroll
        for (int c = 0; c < 3; ++c) {
            jout[c * 3 + 0] = JX[c] + (c == 0 ? 1.0f : 0.0f);
            jout[c * 3 + 1] = JY[c] + (c == 1 ? 1.0f : 0.0f);
            jout[c * 3 + 2] = JZ[c] + (c == 2 ? 1.0f : 0.0f);
        }
    }
}

extern "C" void kernel_launch(void* const* d_in, const int* in_sizes, int n_in,
                              void* d_out, int out_size, void* d_ws, size_t ws_size,
                              hipStream_t stream) {
    const float* points = (const float*)d_in[0];
    const float* grid   = (const float*)d_in[1];
    float*       out    = (float*)d_out;

    int n = in_sizes[0] / 3;
    const int threads = 256;
    int blocks = (n + threads - 1) / threads;
    if (blocks > 1024) blocks = 1024;   // amortize one TDM grid-fill per block

    ffd_eval_kernel<<<blocks, threads, 0, stream>>>(points, grid, out, n);
}